// XYZTTriPlane_29111288332973
// MI455X (gfx1250) — compile-verified
//
#include <hip/hip_runtime.h>
#include <hip/hip_bf16.h>
#include <stdint.h>

#define FEATURES 32
#define RES 512
#define TIME_DIM 16
#define N_PTS 1048576
#define PTS_PER_BLOCK 256
#define THREADS 256

typedef uint32_t u32x4 __attribute__((ext_vector_type(4)));
typedef uint32_t u32x8 __attribute__((ext_vector_type(8)));

// Bilinear grid_sample, align_corners=True. `ga` indexes W, `gb` indexes H.
template <int H, int W>
__device__ __forceinline__ float sample_plane(const float* __restrict__ plane,
                                              int c, float ga, float gb) {
  float ix = (ga + 1.0f) * 0.5f * (float)(W - 1);
  float iy = (gb + 1.0f) * 0.5f * (float)(H - 1);
  float ix0f = fminf(fmaxf(floorf(ix), 0.0f), (float)(W - 1));
  float iy0f = fminf(fmaxf(floorf(iy), 0.0f), (float)(H - 1));
  int ix0 = (int)ix0f;
  int iy0 = (int)iy0f;
  int ix1 = min(ix0 + 1, W - 1);
  int iy1 = min(iy0 + 1, H - 1);
  float wx = ix - ix0f;
  float wy = iy - iy0f;
  const float* p = plane + (size_t)c * (size_t)(H * W);
  float v00 = p[iy0 * W + ix0];
  float v01 = p[iy0 * W + ix1];
  float v10 = p[iy1 * W + ix0];
  float v11 = p[iy1 * W + ix1];
  float mx = 1.0f - wx, my = 1.0f - wy;
  return v00 * (mx * my) + v01 * (wx * my) + v10 * (mx * wy) + v11 * (wx * wy);
}

__global__ __launch_bounds__(THREADS) void triplane_sample_kernel(
    const float* __restrict__ x,
    const float* __restrict__ pxy, const float* __restrict__ pxz,
    const float* __restrict__ pyz, const float* __restrict__ pxt,
    const float* __restrict__ pyt, const float* __restrict__ pzt,
    float* __restrict__ out) {
  __shared__ float4 sc[PTS_PER_BLOCK];  // staged coordinates (4 KB)

  const int tid = threadIdx.x;
  const int p0 = blockIdx.x * PTS_PER_BLOCK;

  // ---- TDM: async DMA the block's coordinate tile (1024 fp32) into LDS ----
  if (tid < 32) {  // wave 0 only (uniform branch; TDM is per-wave, EXEC-ignored)
    uint64_t ga = (uint64_t)(uintptr_t)(x + (size_t)p0 * 4);
    uint32_t lds = (uint32_t)(uintptr_t)(&sc[0]);  // low 32 bits = LDS byte addr
    const uint32_t nelem = PTS_PER_BLOCK * 4;      // fp32 elements in tile
    u32x4 g0;
    g0[0] = 1u;                                    // count=1, user descriptor
    g0[1] = lds;                                   // lds_addr
    g0[2] = (uint32_t)ga;                          // global_addr[31:0]
    g0[3] = ((uint32_t)(ga >> 32) & 0x01FFFFFFu)   // global_addr[56:32]
            | (2u << 30);                          // type = 2 ("image")
    u32x8 g1;
    g1[0] = 2u << 16;                              // data_size = 4 bytes
    g1[1] = (nelem & 0xFFFFu) << 16;               // tensor_dim0[15:0]
    g1[2] = (nelem >> 16) | (1u << 16);            // tensor_dim0[31:16] | tensor_dim1=1
    g1[3] = (nelem & 0xFFFFu) << 16;               // tile_dim0 = nelem
    g1[4] = 0u;                                    // tile_dim1/2 unused
    g1[5] = nelem;                                 // tensor_dim0_stride[31:0]
    g1[6] = 0u;                                    // stride hi / dim1 stride
    g1[7] = 0u;
    asm volatile("tensor_load_to_lds %0, %1" ::"s"(g0), "s"(g1) : "memory");
    __builtin_amdgcn_s_wait_tensorcnt(0);
  }
  __syncthreads();

  const int wave = tid >> 5;
  const int lane = tid & 31;  // lane == channel -> 128B fully-coalesced stores
  const int pts_per_wave = PTS_PER_BLOCK / (THREADS / 32);

  for (int i = 0; i < pts_per_wave; ++i) {
    int pl = wave * pts_per_wave + i;
    float4 cd = sc[pl];  // LDS broadcast (all lanes same address)

    // u = x/2 + 0.5 ; g = 2u - 1  (rounding-exact, matches reference)
    float gx = (cd.x * 0.5f + 0.5f) * 2.0f - 1.0f;
    float gy = (cd.y * 0.5f + 0.5f) * 2.0f - 1.0f;
    float gz = (cd.z * 0.5f + 0.5f) * 2.0f - 1.0f;
    float gt = (cd.w * 0.5f + 0.5f) * 2.0f - 1.0f;

    // Planes are L2-resident (99 MB << 192 MB L2): default (RT) load hints.
    float f_xy = sample_plane<RES, RES>(pxy, lane, gx, gy);
    float f_xz = sample_plane<RES, RES>(pxz, lane, gx, gz);
    float f_yz = sample_plane<RES, RES>(pyz, lane, gy, gz);
    float f_xt = sample_plane<RES, TIME_DIM>(pxt, lane, gx, gt);
    float f_yt = sample_plane<RES, TIME_DIM>(pyt, lane, gy, gt);
    float f_zt = sample_plane<RES, TIME_DIM>(pzt, lane, gz, gt);

    // 805 MB write-once stream: non-temporal so it never evicts the planes.
    float* o = out + (size_t)(p0 + pl) * 192 + lane;
    __builtin_nontemporal_store(f_xy, o + 0);
    __builtin_nontemporal_store(f_xz, o + 32);
    __builtin_nontemporal_store(f_yz, o + 64);
    __builtin_nontemporal_store(f_xt, o + 96);
    __builtin_nontemporal_store(f_yt, o + 128);
    __builtin_nontemporal_store(f_zt, o + 160);
  }
}

extern "C" void kernel_launch(void* const* d_in, const int* in_sizes, int n_in,
                              void* d_out, int out_size, void* d_ws, size_t ws_size,
                              hipStream_t stream) {
  const float* x   = (const float*)d_in[0];
  const float* pxy = (const float*)d_in[1];
  const float* pxz = (const float*)d_in[2];
  const float* pyz = (const float*)d_in[3];
  const float* pxt = (const float*)d_in[4];
  const float* pyt = (const float*)d_in[5];
  const float* pzt = (const float*)d_in[6];
  float* out = (float*)d_out;

  dim3 grid(N_PTS / PTS_PER_BLOCK);
  dim3 block(THREADS);
  triplane_sample_kernel<<<grid, block, 0, stream>>>(x, pxy, pxz, pyz, pxt, pyt,
                                                     pzt, out);
}